// LearnOnlyGaussianPooling_74191265071621
// MI455X (gfx1250) — compile-verified
//
#include <hip/hip_runtime.h>
#include <math.h>

typedef __attribute__((ext_vector_type(2))) float v2f;
typedef __attribute__((ext_vector_type(8))) float v8f;
typedef __attribute__((ext_vector_type(4))) int   v4i;

#define B_ 4
#define N_ 4096
#define C_ 512
#define K_ 683
#define M_ (B_*N_)
#define NEGV (-1e9f)

// output layout (floats), reference tuple order: A, s_c, mu_c, Sigma_c, 4 scalars
#define OFF_SC   ((size_t)M_*K_)                 // 11,190,272
#define OFF_MU   (OFF_SC + (size_t)B_*K_*C_)     // +1,398,784
#define OFF_SIG  (OFF_MU + (size_t)B_*K_*3)
#define OFF_LOSS (OFF_SIG + (size_t)B_*K_*9)

// workspace layout (floats)
#define WS_SPROJ 0
#define WS_DENOM ((size_t)M_*C_)                 // 8,388,608
#define WS_SCAL  (WS_DENOM + (size_t)B_*K_)      // 8 scalar slots

#define WMMA_F32(a, b, c) __builtin_amdgcn_wmma_f32_16x16x4_f32( \
    false, (a), false, (b), (short)0, (c), false, false)

// --- async global->LDS staging (CDNA5 path), with portable fallback --------
// builtin signature (from compiler diagnostic): (v4i* global_src, v4i* lds_dst,
// imm offset, imm cpol) with generic pointers.
#if __has_builtin(__builtin_amdgcn_global_load_async_to_lds_b128) && \
    __has_builtin(__builtin_amdgcn_s_wait_asynccnt)
#define HAVE_ASYNC_LDS 1
#define ASYNC_CP_B128(gp, lp)                                             \
  __builtin_amdgcn_global_load_async_to_lds_b128((v4i*)(gp), (v4i*)(lp), 0, 0)
#else
#define HAVE_ASYNC_LDS 0
#endif

// ---------------------------------------------------------------------------
// Kernel 1: s_proj[m][d] = sum_c s[m][c]*W[d][c] + b[d]
// 64x64 register-blocked WMMA: 16 v8f accumulators, 4 A + 4 B float2 loads
// feed 16 v_wmma per k-step (4x reuse of both operands).
// ---------------------------------------------------------------------------
__global__ __launch_bounds__(256) void k_proj(const float* __restrict__ s,
                                              const float* __restrict__ W,
                                              const float* __restrict__ bias,
                                              float* __restrict__ sproj) {
  const int wave = threadIdx.x >> 5, lane = threadIdx.x & 31;
  const int tile = blockIdx.x * 8 + wave;           // (16384/64)*(512/64)=2048 tiles
  const int NT = C_ / 64;                           // 8
  const int mt = tile / NT, nt = tile % NT;
  const int m0 = mt * 64, n0 = nt * 64;
  const int lm = lane & 15, half = lane >> 4, h2 = half * 2;

  const float* Ar[4];
  const float* Br[4];
#pragma unroll
  for (int i = 0; i < 4; ++i) {
    Ar[i] = s + (size_t)(m0 + i * 16 + lm) * C_ + h2;
    Br[i] = W + (size_t)(n0 + i * 16 + lm) * C_ + h2;
  }
  v8f acc[4][4] = {};
#pragma unroll 1
  for (int kb = 0; kb < C_; kb += 4) {
    v2f a[4], bv[4];
#pragma unroll
    for (int i = 0; i < 4; ++i) a[i] = *(const v2f*)(Ar[i] + kb);
#pragma unroll
    for (int j = 0; j < 4; ++j) bv[j] = *(const v2f*)(Br[j] + kb);
#pragma unroll
    for (int i = 0; i < 4; ++i)
#pragma unroll
      for (int j = 0; j < 4; ++j) acc[i][j] = WMMA_F32(a[i], bv[j], acc[i][j]);
  }
#pragma unroll
  for (int j = 0; j < 4; ++j) {
    const int col = n0 + j * 16 + lm;
    const float bb = bias[col];
#pragma unroll
    for (int i = 0; i < 4; ++i)
#pragma unroll
      for (int r = 0; r < 8; ++r) {
        const int row = m0 + i * 16 + r + 8 * half;  // C/D layout: vgpr r -> M=r / M=8+r
        sproj[(size_t)row * C_ + col] = acc[i][j][r] + bb;
      }
  }
}

// ---------------------------------------------------------------------------
// Kernel 2: logits[m][k] = sum_c s_proj[m][c]*slots[k][c]   (TAU==1)
// Block = 128 rows x 64 slots macro-tile. The 64x512 slots panel (128KB) is
// staged once per block into LDS with async b128 copies (s_wait_asynccnt),
// then all 8 waves stream it from LDS (8x reuse, conflict-free via +4 pad).
// Output written straight into d_out's A region (softmax runs in-place).
// ---------------------------------------------------------------------------
#define LB_SLOTS 64
#define LB_PAD   (C_ + 4)   // row stride 516 floats -> 4-bank lane stride
__global__ __launch_bounds__(256) void k_logits(const float* __restrict__ sproj,
                                                const float* __restrict__ slots,
                                                float* __restrict__ out) {
  __shared__ float Bp[LB_SLOTS][LB_PAD];            // 132KB
  const int NTB = (K_ + 63) / 64;                   // 11 slot-blocks
  const int m0 = (blockIdx.x / NTB) * 128;
  const int slot0 = (blockIdx.x % NTB) * 64;
  const int t = threadIdx.x;

  // stage slots[slot0 .. slot0+63][0..511] -> Bp (clamped rows for last block)
  for (int c = t; c < LB_SLOTS * (C_ / 4); c += 256) {   // 16B chunks
    const int row = c >> 7;                               // /(C_/4)=128
    const int cc = (c & 127) * 4;
    int gr = slot0 + row; if (gr > K_ - 1) gr = K_ - 1;
    const float* gp = slots + (size_t)gr * C_ + cc;
#if HAVE_ASYNC_LDS
    ASYNC_CP_B128(gp, &Bp[row][cc]);
#else
    *(float4*)&Bp[row][cc] = *(const float4*)gp;
#endif
  }
#if HAVE_ASYNC_LDS
  __builtin_amdgcn_s_wait_asynccnt(0);
#endif
  __syncthreads();

  const int wave = t >> 5, lane = t & 31;
  const int lm = lane & 15, half = lane >> 4, h2 = half * 2;
  const int m0w = m0 + wave * 16;
  const float* Arow = sproj + (size_t)(m0w + lm) * C_ + h2;

  v8f acc[4] = {};
#pragma unroll 2
  for (int kb = 0; kb < C_; kb += 4) {
    const v2f a = *(const v2f*)(Arow + kb);
#pragma unroll
    for (int g = 0; g < 4; ++g) {
      const v2f bv = *(const v2f*)&Bp[g * 16 + lm][kb + h2];
      acc[g] = WMMA_F32(a, bv, acc[g]);
    }
  }
#pragma unroll
  for (int g = 0; g < 4; ++g) {
    const int slot = slot0 + g * 16 + lm;
    if (slot < K_) {
#pragma unroll
      for (int r = 0; r < 8; ++r) {
        const int row = m0w + r + 8 * half;
        out[(size_t)row * K_ + slot] = acc[g][r];
      }
    }
  }
}

// ---------------------------------------------------------------------------
__global__ void k_init(float* scal) {
  if (threadIdx.x < 8) scal[threadIdx.x] = 0.0f;
}

// ---------------------------------------------------------------------------
// Kernel 3: in-place masked row softmax over K; accumulates entropy loss.
// ---------------------------------------------------------------------------
__global__ __launch_bounds__(256) void k_softmax(float* __restrict__ A,
                                                 const float* __restrict__ mask,
                                                 float* __restrict__ scal) {
  __shared__ float red[256];
  const int row = blockIdx.x, t = threadIdx.x;
  const float mv = mask[row];
  float* L = A + (size_t)row * K_;

  float vals[3];
  float mx = -INFINITY;
#pragma unroll
  for (int i = 0; i < 3; ++i) {
    const int k = t + i * 256;
    float v = -INFINITY;
    if (k < K_) v = (mv < 0.5f) ? NEGV : L[k];
    vals[i] = v;
    mx = fmaxf(mx, v);
  }
  red[t] = mx; __syncthreads();
  for (int s2 = 128; s2 > 0; s2 >>= 1) { if (t < s2) red[t] = fmaxf(red[t], red[t + s2]); __syncthreads(); }
  mx = red[0]; __syncthreads();

  float sum = 0.0f;
#pragma unroll
  for (int i = 0; i < 3; ++i) {
    const int k = t + i * 256;
    if (k < K_) { vals[i] = expf(vals[i] - mx); sum += vals[i]; }
  }
  red[t] = sum; __syncthreads();
  for (int s2 = 128; s2 > 0; s2 >>= 1) { if (t < s2) red[t] += red[t + s2]; __syncthreads(); }
  const float inv = 1.0f / red[0]; __syncthreads();

  float ent = 0.0f;
#pragma unroll
  for (int i = 0; i < 3; ++i) {
    const int k = t + i * 256;
    if (k < K_) {
      const float a = vals[i] * inv * mv;
      L[k] = a;
      const float ac = fmaxf(a, 1e-8f);
      ent -= ac * logf(ac);
    }
  }
  red[t] = ent; __syncthreads();
  for (int s2 = 128; s2 > 0; s2 >>= 1) { if (t < s2) red[t] += red[t + s2]; __syncthreads(); }
  if (t == 0) { atomicAdd(&scal[0], red[0] * mv); atomicAdd(&scal[1], mv); }
}

// ---------------------------------------------------------------------------
// Kernel 4: denom[b][k] = sum_n A[b][n][k]  (coalesced over k)
// ---------------------------------------------------------------------------
__global__ __launch_bounds__(256) void k_colsum(const float* __restrict__ A,
                                                float* __restrict__ denom) {
  const int b = blockIdx.x / 3;
  const int k = (blockIdx.x % 3) * 256 + threadIdx.x;
  if (k >= K_) return;
  const float* base = A + (size_t)b * N_ * K_ + k;
  float sum = 0.0f;
  for (int n = 0; n < N_; ++n) sum += base[(size_t)n * K_];
  denom[b * K_ + k] = sum;
}

// ---------------------------------------------------------------------------
// Kernel 5: s_c[b][k][c] = (sum_n A[b][n][k]*s[b][n][c]) / max(denom,1e-8)
// 16x64 blocking: each (expensive, stride-683) A load feeds 4 WMMAs.
// ---------------------------------------------------------------------------
__global__ __launch_bounds__(256) void k_sc(const float* A,   // == d_out A region
                                            const float* __restrict__ s,
                                            const float* __restrict__ denom,
                                            float* out) {
  const int wave = threadIdx.x >> 5, lane = threadIdx.x & 31;
  const int tile = blockIdx.x * 8 + wave;           // B_*43*8 = 1376 tiles, exact
  const int MT = (K_ + 15) / 16, NT = C_ / 64;      // 43, 8
  const int b = tile / (MT * NT);
  const int r = tile % (MT * NT);
  const int mt = r / NT, ntg = r % NT;
  const int m0 = mt * 16, n0 = ntg * 64;
  const int lm = lane & 15, half = lane >> 4, h2 = half * 2;

  const int krow = m0 + lm;
  const int kc = krow < K_ ? krow : (K_ - 1);
  const float* Abase = A + (size_t)b * N_ * K_ + kc;
  const float* Sb[4];
#pragma unroll
  for (int j = 0; j < 4; ++j)
    Sb[j] = s + (size_t)b * N_ * C_ + n0 + j * 16 + lm;

  v8f acc[4] = {};
#pragma unroll 1
  for (int nb = 0; nb < N_; nb += 4) {
    const int r0 = nb + h2;
    v2f a;
    a.x = Abase[(size_t)r0 * K_];
    a.y = Abase[(size_t)(r0 + 1) * K_];
#pragma unroll
    for (int j = 0; j < 4; ++j) {
      v2f bv;
      bv.x = Sb[j][(size_t)r0 * C_];
      bv.y = Sb[j][(size_t)(r0 + 1) * C_];
      acc[j] = WMMA_F32(a, bv, acc[j]);
    }
  }
#pragma unroll
  for (int rr = 0; rr < 8; ++rr) {
    const int row = m0 + rr + 8 * half;
    if (row < K_) {
      const float d = fmaxf(denom[b * K_ + row], 1e-8f);
#pragma unroll
      for (int j = 0; j < 4; ++j)
        out[OFF_SC + (size_t)(b * K_ + row) * C_ + n0 + j * 16 + lm] = acc[j][rr] / d;
    }
  }
}

// ---------------------------------------------------------------------------
// Kernel 6: per (b,k): mu_c, Sigma_c single pass:
//   w = A/(S+1e-8), alpha=Σw, mu_c=Σw*mu, M2=Σw*mu*muᵀ, intra=Σw*Sigma
//   Sigma_c = intra + M2 - (2-alpha)*mu_c*mu_cᵀ
// ---------------------------------------------------------------------------
__global__ __launch_bounds__(128) void k_moments(const float* A,
                                                 const float* __restrict__ mu,
                                                 const float* __restrict__ Sig,
                                                 const float* __restrict__ denom,
                                                 float* out) {
  __shared__ float red[19][128];
  const int b = blockIdx.x / K_, k = blockIdx.x % K_;
  const int t = threadIdx.x;
  const float invS = 1.0f / (denom[b * K_ + k] + 1e-8f);
  float acc[19];
#pragma unroll
  for (int j = 0; j < 19; ++j) acc[j] = 0.0f;

  const float* Ab = A + (size_t)b * N_ * K_ + k;
  const float* mb = mu + (size_t)b * N_ * 3;
  const float* Sb = Sig + (size_t)b * N_ * 9;
  for (int n = t; n < N_; n += 128) {
    const float w = Ab[(size_t)n * K_] * invS;
    const float x = mb[n * 3 + 0], y = mb[n * 3 + 1], z = mb[n * 3 + 2];
    acc[0] += w;
    acc[1] += w * x; acc[2] += w * y; acc[3] += w * z;
    acc[4] += w * x * x; acc[5] += w * x * y; acc[6] += w * x * z;
    acc[7] += w * y * y; acc[8] += w * y * z; acc[9] += w * z * z;
#pragma unroll
    for (int j = 0; j < 9; ++j) acc[10 + j] += w * Sb[n * 9 + j];
  }
#pragma unroll
  for (int j = 0; j < 19; ++j) red[j][t] = acc[j];
  __syncthreads();
  for (int s2 = 64; s2 > 0; s2 >>= 1) {
    if (t < s2)
#pragma unroll
      for (int j = 0; j < 19; ++j) red[j][t] += red[j][t + s2];
    __syncthreads();
  }
  if (t == 0) {
    const float alpha = red[0][0];
    const float m3[3] = { red[1][0], red[2][0], red[3][0] };
    out[OFF_MU + (size_t)(b * K_ + k) * 3 + 0] = m3[0];
    out[OFF_MU + (size_t)(b * K_ + k) * 3 + 1] = m3[1];
    out[OFF_MU + (size_t)(b * K_ + k) * 3 + 2] = m3[2];
    const float M2[9] = { red[4][0], red[5][0], red[6][0],
                          red[5][0], red[7][0], red[8][0],
                          red[6][0], red[8][0], red[9][0] };
    const float c2 = 2.0f - alpha;
#pragma unroll
    for (int i = 0; i < 3; ++i)
#pragma unroll
      for (int j = 0; j < 3; ++j)
        out[OFF_SIG + (size_t)(b * K_ + k) * 9 + i * 3 + j] =
            red[10 + i * 3 + j][0] + M2[i * 3 + j] - c2 * m3[i] * m3[j];
  }
}

// ---------------------------------------------------------------------------
// Kernel 7: occupancy loss -> scal[3]
// ---------------------------------------------------------------------------
__global__ __launch_bounds__(1024) void k_occ(const float* __restrict__ denom,
                                              float* __restrict__ scal) {
  __shared__ float red[1024];
  __shared__ float bs[B_];
  const int t = threadIdx.x;
  for (int b = 0; b < B_; ++b) {
    float p = 0.0f;
    for (int k = t; k < K_; k += 1024) p += denom[b * K_ + k];
    red[t] = p; __syncthreads();
    for (int s2 = 512; s2 > 0; s2 >>= 1) { if (t < s2) red[t] += red[t + s2]; __syncthreads(); }
    if (t == 0) bs[b] = fmaxf(red[0], 1e-8f);
    __syncthreads();
  }
  float accv = 0.0f;
  const float invK = 1.0f / (float)K_;
  for (int i = t; i < B_ * K_; i += 1024) {
    const float o = denom[i] / bs[i / K_] - invK;
    accv += o * o;
  }
  red[t] = accv; __syncthreads();
  for (int s2 = 512; s2 > 0; s2 >>= 1) { if (t < s2) red[t] += red[t + s2]; __syncthreads(); }
  if (t == 0) scal[3] = red[0] / (float)(B_ * K_);
}

// ---------------------------------------------------------------------------
// Kernel 8: repulsion loss. One thread per (b,p): sweep q, symmetric 3x3
// adjugate inverse, running top-4 of score=-0.5*dist2, sum relu(top+1).
// ---------------------------------------------------------------------------
__global__ __launch_bounds__(256) void k_rep(const float* __restrict__ muc,
                                             const float* __restrict__ sgc,
                                             float* __restrict__ scal) {
  const int t = blockIdx.x * 256 + threadIdx.x;
  if (t >= B_ * K_) return;
  const int b = t / K_, p = t % K_;
  const float* mu = muc + (size_t)b * K_ * 3;
  const float* Sg = sgc + (size_t)b * K_ * 9;
  const float mpx = mu[p * 3 + 0], mpy = mu[p * 3 + 1], mpz = mu[p * 3 + 2];
  const float s00 = Sg[p * 9 + 0], s01 = Sg[p * 9 + 1], s02 = Sg[p * 9 + 2];
  const float s11 = Sg[p * 9 + 4], s12 = Sg[p * 9 + 5], s22 = Sg[p * 9 + 8];

  float top[4] = { NEGV, NEGV, NEGV, NEGV };
  for (int q = 0; q < K_; ++q) {
    if (q == p) continue;
    const float dx = mpx - mu[q * 3 + 0];
    const float dy = mpy - mu[q * 3 + 1];
    const float dz = mpz - mu[q * 3 + 2];
    const float m00 = s00 + Sg[q * 9 + 0] + 1e-6f;
    const float m01 = s01 + Sg[q * 9 + 1];
    const float m02 = s02 + Sg[q * 9 + 2];
    const float m11 = s11 + Sg[q * 9 + 4] + 1e-6f;
    const float m12 = s12 + Sg[q * 9 + 5];
    const float m22 = s22 + Sg[q * 9 + 8] + 1e-6f;
    const float c00 = m11 * m22 - m12 * m12;
    const float c01 = m02 * m12 - m01 * m22;
    const float c02 = m01 * m12 - m02 * m11;
    const float det = m00 * c00 + m01 * c01 + m02 * c02;
    const float id = 1.0f / det;
    const float i00 = c00 * id, i01 = c01 * id, i02 = c02 * id;
    const float i11 = (m00 * m22 - m02 * m02) * id;
    const float i12 = (m01 * m02 - m00 * m12) * id;
    const float i22 = (m00 * m11 - m01 * m01) * id;
    const float d2 = i00 * dx * dx + i11 * dy * dy + i22 * dz * dz +
                     2.0f * (i01 * dx * dy + i02 * dx * dz + i12 * dy * dz);
    const float sc = -0.5f * d2;
    if (sc > top[3]) {
      top[3] = sc;
      if (top[3] > top[2]) { float tt = top[2]; top[2] = top[3]; top[3] = tt;
        if (top[2] > top[1]) { tt = top[1]; top[1] = top[2]; top[2] = tt;
          if (top[1] > top[0]) { tt = top[0]; top[0] = top[1]; top[1] = tt; } } }
    }
  }
  float loc = 0.0f;
#pragma unroll
  for (int i = 0; i < 4; ++i) loc += fmaxf(top[i] + 1.0f, 0.0f);  // relu(top - (-1))
  atomicAdd(&scal[2], loc);
}

// ---------------------------------------------------------------------------
__global__ void k_final(const float* __restrict__ scal, float* __restrict__ out) {
  if (threadIdx.x == 0) {
    const float occ = scal[3];
    const float rep = scal[2] / (float)(B_ * K_ * 4);
    const float ent = scal[0] / fmaxf(scal[1], 1.0f);
    out[OFF_LOSS + 0] = occ;
    out[OFF_LOSS + 1] = rep;
    out[OFF_LOSS + 2] = ent;
    out[OFF_LOSS + 3] = 1.0f * occ + 0.1f * rep + 0.01f * ent;
  }
}

// ---------------------------------------------------------------------------
extern "C" void kernel_launch(void* const* d_in, const int* in_sizes, int n_in,
                              void* d_out, int out_size, void* d_ws, size_t ws_size,
                              hipStream_t stream) {
  const float* s     = (const float*)d_in[0];
  const float* mu    = (const float*)d_in[1];
  const float* Sig   = (const float*)d_in[2];
  const float* mask  = (const float*)d_in[3];
  const float* W     = (const float*)d_in[4];
  const float* bias  = (const float*)d_in[5];
  const float* slots = (const float*)d_in[6];
  float* out = (float*)d_out;
  float* ws  = (float*)d_ws;
  float* sproj = ws + WS_SPROJ;
  float* denom = ws + WS_DENOM;
  float* scal  = ws + WS_SCAL;

  k_init   <<<1, 32, 0, stream>>>(scal);
  k_proj   <<<256, 256, 0, stream>>>(s, W, bias, sproj);        // 2048 64x64 tiles
  k_logits <<<(M_ / 128) * 11, 256, 0, stream>>>(sproj, slots, out); // 1408 blocks
  k_softmax<<<M_, 256, 0, stream>>>(out, mask, scal);
  k_colsum <<<B_ * 3, 256, 0, stream>>>(out, denom);
  k_sc     <<<172, 256, 0, stream>>>(out, s, denom, out);       // 1376 16x64 tiles
  k_moments<<<B_ * K_, 128, 0, stream>>>(out, mu, Sig, denom, out);
  k_occ    <<<1, 1024, 0, stream>>>(denom, scal);
  k_rep    <<<(B_ * K_ + 255) / 256, 256, 0, stream>>>(out + OFF_MU, out + OFF_SIG, scal);
  k_final  <<<1, 32, 0, stream>>>(scal, out);
}